// DySepConvAtten_13408887899069
// MI455X (gfx1250) — compile-verified
//
#include <hip/hip_runtime.h>
#include <hip/hip_bf16.h>

#define BB  128
#define NN  512
#define CC  512
#define KW  3
#define LNEPS 1e-5f

typedef __attribute__((ext_vector_type(2))) float v2f;
typedef __attribute__((ext_vector_type(8))) float v8f;
typedef __attribute__((ext_vector_type(4))) unsigned int v4u;
typedef __attribute__((ext_vector_type(8))) int v8i;
typedef __attribute__((ext_vector_type(4))) int v4i;

// ---------------------------------------------------------------------------
// LDS byte offset of a __shared__ object (workgroup-relative, AS(3) ptrtoint)
// ---------------------------------------------------------------------------
__device__ __forceinline__ unsigned lds_addr_of(const void* p) {
  return (unsigned)(unsigned long long)
      (__attribute__((address_space(3))) const void*)p;
}

// ---------------------------------------------------------------------------
// TDM: 2D tile load Global -> LDS (Tensor Data Mover, TENSOR_LOAD_TO_LDS).
// Descriptor packing per CDNA5 ISA D# group0/group1 (CDNA5 ISA).
//   tile_d0 = contiguous elements per row, tile_d1 = rows,
//   row_stride = elements between rows. data_size = 4 bytes (f32).
// Tensor dims are set equal to tile dims (tiles are fully in-bounds).
// This toolchain exposes the 6-arg builtin:
//   (v4u g0, v8i g1, v4i g2, v4i g3, v8i g4, i32 cpol)
// ---------------------------------------------------------------------------
__device__ __forceinline__ void tdm_load_2d(const void* lds_dst,
                                            const float* gsrc,
                                            unsigned tile_d0, unsigned tile_d1,
                                            unsigned row_stride)
{
  unsigned long long ga = (unsigned long long)gsrc;
  v4u g0;
  g0[0] = 1u;                                  // count=1 (valid), user, load
  g0[1] = lds_addr_of(lds_dst);                // lds_addr (bytes)
  g0[2] = (unsigned)ga;                        // global_addr[31:0]
  g0[3] = (unsigned)((ga >> 32) & 0x01FFFFFFu) // global_addr[56:32]
        | (2u << 30);                          // type = 2 ("image")
  v8i g1;
  g1[0] = (int)(2u << 16);                                 // data_size = 4B
  g1[1] = (int)((tile_d0 & 0xFFFFu) << 16);                // tensor_dim0 lo16
  g1[2] = (int)((tile_d0 >> 16) | ((tile_d1 & 0xFFFFu) << 16)); // d0 hi | d1 lo
  g1[3] = (int)((tile_d1 >> 16) | ((tile_d0 & 0xFFFFu) << 16)); // d1 hi | tile_dim0
  g1[4] = (int)(tile_d1 & 0xFFFFu);                        // tile_dim1 | tile_dim2=0
  g1[5] = (int)row_stride;                                 // tensor_dim0_stride lo32
  g1[6] = 0;                                               // stride hi | dim1_stride lo
  g1[7] = 0;
  v4i z4 = {0, 0, 0, 0};                                   // groups 2/3 unused (2D)
  v8i z8 = {0, 0, 0, 0, 0, 0, 0, 0};
  __builtin_amdgcn_tensor_load_to_lds(g0, g1, z4, z4, z8, 0);
}

// ---------------------------------------------------------------------------
// Kernel 1: dw[b,n,0:3] = query[b,n,:] @ W_lin[:,0:3] + b_lin[0:3]
// One wave per output row; shuffle reduction. dw stored with stride 4.
// ---------------------------------------------------------------------------
__global__ __launch_bounds__(256) void dw_kernel(
    const float* __restrict__ q, const float* __restrict__ W,
    const float* __restrict__ bl, float* __restrict__ dw)
{
  int row  = blockIdx.x * 8 + (threadIdx.x >> 5);
  int lane = threadIdx.x & 31;
  const float* qr = q + (size_t)row * CC;
  float a0 = 0.f, a1 = 0.f, a2 = 0.f;
  for (int c = lane; c < CC; c += 32) {
    float qv = qr[c];
    const float* wr = W + (size_t)c * (NN + KW);
    a0 = fmaf(qv, wr[0], a0);
    a1 = fmaf(qv, wr[1], a1);
    a2 = fmaf(qv, wr[2], a2);
  }
  for (int off = 16; off > 0; off >>= 1) {
    a0 += __shfl_down(a0, off, 32);
    a1 += __shfl_down(a1, off, 32);
    a2 += __shfl_down(a2, off, 32);
  }
  if (lane == 0) {
    float* o = dw + (size_t)row * 4;
    o[0] = a0 + bl[0];
    o[1] = a1 + bl[1];
    o[2] = a2 + bl[2];
  }
}

// ---------------------------------------------------------------------------
// Kernel 2: pw[(b,n), j] = query[(b,n), :] @ W_lin[:, 3+j] + b_lin[3+j]
// 16 rows x 512 cols per workgroup; 8 waves x (16x64). WMMA f32 16x16x4.
// Tiles staged by TDM, double-buffered, pipelined with s_wait_tensorcnt.
// ---------------------------------------------------------------------------
__global__ __launch_bounds__(256) void pw_gemm_kernel(
    const float* __restrict__ q, const float* __restrict__ W,
    const float* __restrict__ bl, float* __restrict__ pw)
{
  __shared__ float At[2][16][16];
  __shared__ float Bt[2][16][512];

  const int tid  = threadIdx.x;
  const int wave = tid >> 5;
  const int lane = tid & 31;
  const int half = lane >> 4;   // 0: lanes 0-15, 1: lanes 16-31
  const int lm   = lane & 15;
  const int r0   = blockIdx.x * 16;

  const float* Abase = q + (size_t)r0 * CC;   // 16 rows, row stride CC
  const float* Bbase = W + KW;                // cols [3,515), row stride 515

  v8f acc[4];
  #pragma unroll
  for (int t = 0; t < 4; ++t)
    #pragma unroll
    for (int j = 0; j < 8; ++j) acc[t][j] = 0.f;

  if (tid < 32) {   // wave 0 drives the TDM (EXEC ignored; one issue per wave)
    tdm_load_2d(&At[0][0][0], Abase, 16, 16, CC);
    tdm_load_2d(&Bt[0][0][0], Bbase, 512, 16, NN + KW);
  }

  int buf = 0;
  for (int k0 = 0; k0 < CC; k0 += 16, buf ^= 1) {
    if (tid < 32) {
      if (k0 + 16 < CC) {
        tdm_load_2d(&At[buf ^ 1][0][0], Abase + (k0 + 16), 16, 16, CC);
        tdm_load_2d(&Bt[buf ^ 1][0][0],
                    Bbase + (size_t)(k0 + 16) * (NN + KW), 512, 16, NN + KW);
        __builtin_amdgcn_s_wait_tensorcnt(2);  // current chunk's 2 ops retired
      } else {
        __builtin_amdgcn_s_wait_tensorcnt(0);  // drain everything
      }
    }
    __syncthreads();

    #pragma unroll
    for (int kk = 0; kk < 16; kk += 4) {
      // A frag (16x4 f32): lanes 0-15 -> K0,K1 ; lanes 16-31 -> K2,K3
      v2f a;
      a[0] = At[buf][lm][kk + 2 * half];
      a[1] = At[buf][lm][kk + 2 * half + 1];
      #pragma unroll
      for (int t = 0; t < 4; ++t) {
        int n = (wave << 6) + (t << 4) + lm;
        // B frag (4x16 f32): VGPR v, half h -> row k = kk + 2*h + v
        v2f b;
        b[0] = Bt[buf][kk + 2 * half][n];
        b[1] = Bt[buf][kk + 2 * half + 1][n];
        acc[t] = __builtin_amdgcn_wmma_f32_16x16x4_f32(
            false, a, false, b, (short)0, acc[t], false, false);
      }
    }
    __syncthreads();   // all reads of buf done before TDM refills it
  }

  // Epilogue: C/D layout VGPR j -> row j + 8*half, col = lane%16 within tile
  #pragma unroll
  for (int t = 0; t < 4; ++t) {
    int col   = (wave << 6) + (t << 4) + lm;
    float bia = bl[KW + col];
    #pragma unroll
    for (int j = 0; j < 8; ++j) {
      int ml = j + 8 * half;
      pw[(size_t)(r0 + ml) * NN + col] = acc[t][j] + bia;
    }
  }
}

// ---------------------------------------------------------------------------
// Kernel 3: depth = relu(dynamic 3-tap conv of value along C), elementwise
// ---------------------------------------------------------------------------
__global__ __launch_bounds__(256) void depth_kernel(
    const float* __restrict__ value, const float* __restrict__ dw,
    float* __restrict__ depth)
{
  size_t idx = (size_t)blockIdx.x * 256 + threadIdx.x;  // over B*N*C
  int    c   = (int)(idx & (CC - 1));
  size_t row = idx >> 9;
  const float* dr = dw + row * 4;
  const float* v  = value + (row << 9);
  float x = dr[1] * v[c];
  if (c > 0)      x = fmaf(dr[0], v[c - 1], x);
  if (c < CC - 1) x = fmaf(dr[2], v[c + 1], x);
  depth[idx] = fmaxf(x, 0.f);
}

// ---------------------------------------------------------------------------
// Kernel 4: out[b,m,:] = LayerNorm( pw[b,m,:] @ depth[b,:,:] ) * gamma + beta
// Same TDM+WMMA skeleton; full 512-col row kept in LDS so LN fuses on-chip.
// ---------------------------------------------------------------------------
__global__ __launch_bounds__(256) void out_gemm_ln_kernel(
    const float* __restrict__ pw, const float* __restrict__ depth,
    const float* __restrict__ gamma, const float* __restrict__ beta,
    float* __restrict__ out)
{
  __shared__ float At[2][16][16];
  __shared__ float Bt[2][16][512];

  const int tid  = threadIdx.x;
  const int wave = tid >> 5;
  const int lane = tid & 31;
  const int half = lane >> 4;
  const int lm   = lane & 15;
  const int b    = blockIdx.x >> 5;          // batch
  const int r0   = (blockIdx.x & 31) << 4;   // row block within batch

  const float* pwB = pw    + (size_t)b * NN * NN;
  const float* dB  = depth + (size_t)b * NN * CC;
  const float* Abase = pwB + (size_t)r0 * NN;

  v8f acc[4];
  #pragma unroll
  for (int t = 0; t < 4; ++t)
    #pragma unroll
    for (int j = 0; j < 8; ++j) acc[t][j] = 0.f;

  if (tid < 32) {
    tdm_load_2d(&At[0][0][0], Abase, 16, 16, NN);
    tdm_load_2d(&Bt[0][0][0], dB, 512, 16, CC);
  }

  int buf = 0;
  for (int k0 = 0; k0 < NN; k0 += 16, buf ^= 1) {
    if (tid < 32) {
      if (k0 + 16 < NN) {
        tdm_load_2d(&At[buf ^ 1][0][0], Abase + (k0 + 16), 16, 16, NN);
        tdm_load_2d(&Bt[buf ^ 1][0][0], dB + (size_t)(k0 + 16) * CC, 512, 16, CC);
        __builtin_amdgcn_s_wait_tensorcnt(2);
      } else {
        __builtin_amdgcn_s_wait_tensorcnt(0);
      }
    }
    __syncthreads();

    #pragma unroll
    for (int kk = 0; kk < 16; kk += 4) {
      v2f a;
      a[0] = At[buf][lm][kk + 2 * half];
      a[1] = At[buf][lm][kk + 2 * half + 1];
      #pragma unroll
      for (int t = 0; t < 4; ++t) {
        int n = (wave << 6) + (t << 4) + lm;
        v2f bf;
        bf[0] = Bt[buf][kk + 2 * half][n];
        bf[1] = Bt[buf][kk + 2 * half + 1][n];
        acc[t] = __builtin_amdgcn_wmma_f32_16x16x4_f32(
            false, a, false, bf, (short)0, acc[t], false, false);
      }
    }
    __syncthreads();
  }

  // Spill accumulators to LDS so each row's full 512 cols are visible for LN.
  // All TDM ops drained (tensorcnt==0), so Bt[0] is safe to reuse.
  #pragma unroll
  for (int t = 0; t < 4; ++t) {
    int col = (wave << 6) + (t << 4) + lm;
    #pragma unroll
    for (int j = 0; j < 8; ++j)
      Bt[0][j + 8 * half][col] = acc[t][j];
  }
  __syncthreads();

  // LayerNorm: 16 threads per row, each owns 32 cols; xor-reduce over 16 lanes
  const int row = tid >> 4;
  const int seg = tid & 15;
  float s = 0.f, ss = 0.f;
  #pragma unroll
  for (int i = 0; i < 32; ++i) {
    float x = Bt[0][row][(seg << 5) + i];
    s  += x;
    ss = fmaf(x, x, ss);
  }
  #pragma unroll
  for (int off = 8; off > 0; off >>= 1) {
    s  += __shfl_xor(s, off, 16);
    ss += __shfl_xor(ss, off, 16);
  }
  float mu  = s * (1.f / (float)CC);
  float var = ss * (1.f / (float)CC) - mu * mu;
  float rs  = rsqrtf(var + LNEPS);

  float* orow = out + ((size_t)b * NN + (r0 + row)) * CC;
  for (int i = 0; i < 32; ++i) {
    int c = (seg << 5) + i;
    float x = (Bt[0][row][c] - mu) * rs;
    orow[c] = fmaf(x, gamma[c], beta[c]);
  }
}

// ---------------------------------------------------------------------------
extern "C" void kernel_launch(void* const* d_in, const int* in_sizes, int n_in,
                              void* d_out, int out_size, void* d_ws, size_t ws_size,
                              hipStream_t stream) {
  const float* q     = (const float*)d_in[0];   // (B,N,C)
  const float* value = (const float*)d_in[1];   // (B,N,C)
  const float* W     = (const float*)d_in[2];   // (C, N+K)
  const float* bl    = (const float*)d_in[3];   // (N+K,)
  const float* gamma = (const float*)d_in[4];   // (C,)
  const float* beta  = (const float*)d_in[5];   // (C,)
  float* out = (float*)d_out;

  // Workspace layout (floats): dw (B*N*4) | pw (B*N*N) | depth (B*N*C)
  float* dw    = (float*)d_ws;
  float* pw    = dw + (size_t)BB * NN * 4;
  float* depth = pw + (size_t)BB * NN * NN;

  dw_kernel      <<<BB * NN / 8,  256, 0, stream>>>(q, W, bl, dw);
  pw_gemm_kernel <<<BB * NN / 16, 256, 0, stream>>>(q, W, bl, pw);
  depth_kernel   <<<BB * NN * (CC / 256), 256, 0, stream>>>(value, dw, depth);
  out_gemm_ln_kernel<<<BB * (NN / 16), 256, 0, stream>>>(pw, depth, gamma, beta, out);
}